// LocallyConnectedNet_901943132206
// MI455X (gfx1250) — compile-verified
//
#include <hip/hip_runtime.h>

// FP32 WMMA path: V_WMMA_F32_16X16X4_F32  (D = A(16x4) * B(4x16) + C(16x16), f32)
// f32 matches the float32 reference in precision class; at ~6.2 GFLOP total the
// net is bandwidth/latency bound on MI455X (~75 MB traffic ~ 3 us at 23.3 TB/s),
// so the f32 matrix path is the right choice.
//
// M-tile = 32 batch rows per wave (two A fragments share every B fragment) to
// double arithmetic intensity. All operands except LC1's gathered A are aligned v2f b64 loads with
// constant IOFFSETs; weights are repacked once into fragment-native layouts.
typedef __attribute__((ext_vector_type(2))) float v2f;
typedef __attribute__((ext_vector_type(8))) float v8f;

static __device__ __forceinline__ v8f wmma4(v2f a, v2f b, v8f c) {
  // 8 args: (neg_a, A, neg_b, B, c_mod, C, reuse_a, reuse_b)
  return __builtin_amdgcn_wmma_f32_16x16x4_f32(false, a, false, b, (short)0, c,
                                               false, false);
}

// ---------------------------------------------------------------------------
// Weight repack prologue kernels (fragment-native layouts, zero-padded)
// ---------------------------------------------------------------------------
// w1:(16,1,9,26,26) -> w1p[pos][o][row], row 0..11 (K padded 9->12, zeros)
__global__ __launch_bounds__(256) void k_pack_w1(const float* __restrict__ w1,
                                                 float* __restrict__ w1p) {
  const int idx = blockIdx.x * 256 + threadIdx.x;  // 676*16*12 = 129792
  if (idx >= 676 * 16 * 12) return;
  const int row = idx % 12;
  const int t = idx / 12;
  const int o = t % 16, pos = t / 16;
  w1p[idx] = (row < 9) ? w1[(o * 9 + row) * 676 + pos] : 0.0f;
}

// w2:(32,16,9,11,11) -> w2p[pos][o][kk'], kk' = k_spatial*16 + c
__global__ __launch_bounds__(256) void k_pack_w2(const float* __restrict__ w2,
                                                 float* __restrict__ w2p) {
  const int idx = blockIdx.x * 256 + threadIdx.x;  // 121*32*144 = 557568
  if (idx >= 121 * 32 * 144) return;
  const int kk = idx % 144;
  const int t = idx / 144;
  const int o = t % 32, pos = t / 32;
  const int k = kk >> 4, c = kk & 15;
  w2p[idx] = w2[((o * 16 + c) * 9 + k) * 121 + pos];
}

// fc1_w:(800,128) -> fc1T[c][k]  (transpose)
__global__ __launch_bounds__(256) void k_pack_fc1(const float* __restrict__ w,
                                                  float* __restrict__ wT) {
  const int idx = blockIdx.x * 256 + threadIdx.x;  // 128*800 = 102400
  if (idx >= 128 * 800) return;
  const int k = idx % 800, c = idx / 800;
  wT[idx] = w[k * 128 + c];
}

// fc2_w:(128,10) -> fc2p[o][k], o padded 10->16 with zeros
__global__ __launch_bounds__(256) void k_pack_fc2(const float* __restrict__ w,
                                                  float* __restrict__ wp) {
  const int idx = blockIdx.x * 256 + threadIdx.x;  // 16*128 = 2048
  if (idx >= 16 * 128) return;
  const int k = idx % 128, o = idx / 128;
  wp[idx] = (o < 10) ? w[k * 10 + o] : 0.0f;
}

// ---------------------------------------------------------------------------
// Kernel 1: LC-conv1 (k=3, C=1) + bias + ReLU + 2x2 maxpool
// x:(4096,1,28,28) w1p(packed) b1:(16,26,26) -> h1 NHWC:(4096,13,13,16)
// One wave = 32-batch x 16-channel tile at one pooled position.
// ---------------------------------------------------------------------------
__global__ __launch_bounds__(256) void k_lc1(const float* __restrict__ x,
                                             const float* __restrict__ w1p,
                                             const float* __restrict__ b1,
                                             float* __restrict__ h1) {
  const int lane = threadIdx.x & 31;
  const int wave = threadIdx.x >> 5;
  const int tile = blockIdx.x * 8 + wave;  // 169 positions * 128 batch tiles
  const int m = lane & 15, kh = lane >> 4;
  const int bt = tile / 169, pos = tile % 169;
  const int pi = pos / 13, pj = pos % 13;
  const int n0 = bt * 32;

  const float* pxA = x + (n0 + m) * 784;        // rows 0..15
  const float* pxB = pxA + 16 * 784;            // rows 16..31

  v8f mxA = {}, mxB = {};  // ReLU(max(...)) == max with 0-init
#pragma unroll
  for (int ci2 = 0; ci2 < 2; ++ci2) {
#pragma unroll
    for (int cj2 = 0; cj2 < 2; ++cj2) {
      const int ci = 2 * pi + ci2, cj = 2 * pj + cj2;  // conv pos 0..25
      const int po = ci * 28 + cj;
      const float* pw = w1p + ((ci * 26 + cj) * 16 + m) * 12 + kh * 2;
      v8f aA = {}, aB = {};
      {  // s=0: k = {0,1} | {2,3};  patch offs (k/3)*28+k%3: 0,1 | 2,28
        const v2f b = *reinterpret_cast<const v2f*>(pw);
        v2f a0, a1;
        const int ox = po + (kh ? 2 : 0), oy = po + (kh ? 28 : 1);
        a0.x = pxA[ox]; a0.y = pxA[oy];
        a1.x = pxB[ox]; a1.y = pxB[oy];
        aA = wmma4(a0, b, aA);
        aB = wmma4(a1, b, aB);
      }
      {  // s=1: k = {4,5} | {6,7};  offs 29,30 | 56,57 (contiguous pairs)
        const v2f b = *reinterpret_cast<const v2f*>(pw + 4);
        const int ob = po + (kh ? 56 : 29);
        const v2f a0 = *reinterpret_cast<const v2f*>(pxA + ob);
        const v2f a1 = *reinterpret_cast<const v2f*>(pxB + ob);
        aA = wmma4(a0, b, aA);
        aB = wmma4(a1, b, aB);
      }
      {  // s=2: k = {8,9} | {10,11}; only k=8 valid (w1p rows 9..11 are 0)
        const v2f b = *reinterpret_cast<const v2f*>(pw + 8);
        v2f a0, a1;
        const float ax0 = pxA[po + 58];  // off(8) = 2*28+2
        const float ax1 = pxB[po + 58];
        a0.x = kh ? 0.0f : ax0; a0.y = 0.0f;
        a1.x = kh ? 0.0f : ax1; a1.y = 0.0f;
        aA = wmma4(a0, b, aA);
        aB = wmma4(a1, b, aB);
      }
      const float bias = b1[(m * 26 + ci) * 26 + cj];  // o = lane&15
#pragma unroll
      for (int r = 0; r < 8; ++r) {
        mxA[r] = fmaxf(mxA[r], aA[r] + bias);
        mxB[r] = fmaxf(mxB[r], aB[r] + bias);
      }
    }
  }
  // NHWC store: lanes 0..15 write 16 contiguous channels
#pragma unroll
  for (int r = 0; r < 8; ++r) {
    h1[(((n0 + r + 8 * kh) * 13 + pi) * 13 + pj) * 16 + m]      = mxA[r];
    h1[(((n0 + 16 + r + 8 * kh) * 13 + pi) * 13 + pj) * 16 + m] = mxB[r];
  }
}

// ---------------------------------------------------------------------------
// Kernel 2: LC-conv2 (k=3, C=16) + bias + ReLU + maxpool
// h1 NHWC:(4096,13,13,16) w2p(packed) b2:(32,11,11) -> h2:(4096,800)
// One wave = 32-batch x 32-channel tile at one pooled position.
// K reordered as kk' = k_spatial*16 + c. Per step: 4 b64 loads -> 4 WMMAs.
// ---------------------------------------------------------------------------
__global__ __launch_bounds__(256) void k_lc2(const float* __restrict__ h1,
                                             const float* __restrict__ w2p,
                                             const float* __restrict__ b2,
                                             float* __restrict__ h2) {
  const int lane = threadIdx.x & 31;
  const int wave = threadIdx.x >> 5;
  const int tile = blockIdx.x * 8 + wave;  // 25 pooled pos * 128 batch tiles
  const int m = lane & 15, kh = lane >> 4;
  const int bt = tile / 25, pos = tile % 25;
  const int pi = pos / 5, pj = pos % 5;
  const int n0 = bt * 32;

  const float* paA = h1 + (n0 + m) * 2704 + kh * 2;  // 169*16 per image
  const float* paB = paA + 16 * 2704;

  v8f mx00 = {}, mx01 = {}, mx10 = {}, mx11 = {};
#pragma unroll
  for (int ci2 = 0; ci2 < 2; ++ci2) {
#pragma unroll
    for (int cj2 = 0; cj2 < 2; ++cj2) {
      const int ci = 2 * pi + ci2, cj = 2 * pj + cj2;  // 0..9 (<11)
      const float* pAa = paA + (ci * 13 + cj) * 16;
      const float* pAb = paB + (ci * 13 + cj) * 16;
      const float* pB0 = w2p + ((ci * 11 + cj) * 32 + m) * 144 + kh * 2;
      const float* pB1 = pB0 + 16 * 144;
      v8f a00 = {}, a01 = {}, a10 = {}, a11 = {};
#pragma unroll
      for (int s = 0; s < 36; ++s) {  // K = 144 exactly
        const int t = s >> 2;         // spatial k 0..8
        const int soff = ((t / 3) * 13 + (t % 3)) * 16;  // compile-time
        const int c0 = (s & 3) * 4;                      // channel base
        const v2f a0 = *reinterpret_cast<const v2f*>(pAa + soff + c0);
        const v2f a1 = *reinterpret_cast<const v2f*>(pAb + soff + c0);
        const v2f b0 = *reinterpret_cast<const v2f*>(pB0 + s * 4);
        const v2f b1 = *reinterpret_cast<const v2f*>(pB1 + s * 4);
        a00 = wmma4(a0, b0, a00);
        a01 = wmma4(a0, b1, a01);
        a10 = wmma4(a1, b0, a10);
        a11 = wmma4(a1, b1, a11);
      }
      const float bias0 = b2[(m * 11 + ci) * 11 + cj];
      const float bias1 = b2[((m + 16) * 11 + ci) * 11 + cj];
#pragma unroll
      for (int r = 0; r < 8; ++r) {
        mx00[r] = fmaxf(mx00[r], a00[r] + bias0);
        mx01[r] = fmaxf(mx01[r], a01[r] + bias1);
        mx10[r] = fmaxf(mx10[r], a10[r] + bias0);
        mx11[r] = fmaxf(mx11[r], a11[r] + bias1);
      }
    }
  }
  // flatten as reference (32,5,5): f = (o*5+pi)*5+pj
  const int f0 = (m * 5 + pi) * 5 + pj;
  const int f1 = ((m + 16) * 5 + pi) * 5 + pj;
#pragma unroll
  for (int r = 0; r < 8; ++r) {
    const int na = n0 + r + 8 * kh;
    const int nb = na + 16;
    h2[na * 800 + f0] = mx00[r];
    h2[na * 800 + f1] = mx01[r];
    h2[nb * 800 + f0] = mx10[r];
    h2[nb * 800 + f1] = mx11[r];
  }
}

// ---------------------------------------------------------------------------
// Kernel 3: FC1  (4096x800)x(800x128) + bias, ReLU  -> h3:(4096,128)
// One wave = 32-batch x 16-col tile; B fragment shared by two A fragments.
// ---------------------------------------------------------------------------
__global__ __launch_bounds__(256) void k_fc1(const float* __restrict__ h2,
                                             const float* __restrict__ wT,
                                             const float* __restrict__ bias,
                                             float* __restrict__ h3) {
  const int lane = threadIdx.x & 31;
  const int wave = threadIdx.x >> 5;
  const int tile = blockIdx.x * 8 + wave;  // 128 batch tiles * 8 col tiles
  const int m = lane & 15, kh = lane >> 4;
  const int bt = tile >> 3, ct = tile & 7;
  const int n0 = bt * 32, c0 = ct * 16;

  const float* pA0 = h2 + (n0 + m) * 800 + kh * 2;
  const float* pA1 = pA0 + 16 * 800;
  const float* pB = wT + (c0 + m) * 800 + kh * 2;

  v8f acc0 = {}, acc1 = {};
  for (int so = 0; so < 25; ++so) {  // 200 K-steps, unroll 8, bump pointers
#pragma unroll
    for (int u = 0; u < 8; ++u) {
      const v2f a0 = *reinterpret_cast<const v2f*>(pA0 + u * 4);
      const v2f a1 = *reinterpret_cast<const v2f*>(pA1 + u * 4);
      const v2f b  = *reinterpret_cast<const v2f*>(pB + u * 4);
      acc0 = wmma4(a0, b, acc0);
      acc1 = wmma4(a1, b, acc1);
    }
    pA0 += 32;
    pA1 += 32;
    pB += 32;
  }
  const float bb = bias[c0 + m];
#pragma unroll
  for (int r = 0; r < 8; ++r) {
    h3[(n0 + r + 8 * kh) * 128 + c0 + m]      = fmaxf(acc0[r] + bb, 0.0f);
    h3[(n0 + 16 + r + 8 * kh) * 128 + c0 + m] = fmaxf(acc1[r] + bb, 0.0f);
  }
}

// ---------------------------------------------------------------------------
// Kernel 4: FC2  (4096x128)x(128x10) + bias  (N padded 10->16, zeros prebaked)
// ---------------------------------------------------------------------------
__global__ __launch_bounds__(256) void k_fc2(const float* __restrict__ h3,
                                             const float* __restrict__ wp,
                                             const float* __restrict__ bias,
                                             float* __restrict__ out) {
  const int lane = threadIdx.x & 31;
  const int wave = threadIdx.x >> 5;
  const int tile = blockIdx.x * 8 + wave;  // 128 batch tiles
  const int m = lane & 15, kh = lane >> 4;
  const int n0 = tile * 32;

  const float* pA0 = h3 + (n0 + m) * 128 + kh * 2;
  const float* pA1 = pA0 + 16 * 128;
  const float* pB = wp + m * 128 + kh * 2;

  v8f acc0 = {}, acc1 = {};
#pragma unroll
  for (int s = 0; s < 32; ++s) {  // K = 128
    const v2f a0 = *reinterpret_cast<const v2f*>(pA0 + s * 4);
    const v2f a1 = *reinterpret_cast<const v2f*>(pA1 + s * 4);
    const v2f b  = *reinterpret_cast<const v2f*>(pB + s * 4);
    acc0 = wmma4(a0, b, acc0);
    acc1 = wmma4(a1, b, acc1);
  }
  if (m < 10) {
    const float bb = bias[m];
#pragma unroll
    for (int r = 0; r < 8; ++r) {
      out[(n0 + r + 8 * kh) * 10 + m]      = acc0[r] + bb;
      out[(n0 + 16 + r + 8 * kh) * 10 + m] = acc1[r] + bb;
    }
  }
}

// ---------------------------------------------------------------------------
extern "C" void kernel_launch(void* const* d_in, const int* in_sizes, int n_in,
                              void* d_out, int out_size, void* d_ws,
                              size_t ws_size, hipStream_t stream) {
  const float* x     = (const float*)d_in[0];
  const float* w1    = (const float*)d_in[1];
  const float* b1    = (const float*)d_in[2];
  const float* w2    = (const float*)d_in[3];
  const float* b2    = (const float*)d_in[4];
  const float* fc1_w = (const float*)d_in[5];
  const float* fc1_b = (const float*)d_in[6];
  const float* fc2_w = (const float*)d_in[7];
  const float* fc2_b = (const float*)d_in[8];

  float* h1  = (float*)d_ws;                   // 4096*13*13*16 NHWC
  float* h2  = h1 + (size_t)4096 * 169 * 16;   // 4096*800
  float* h3  = h2 + (size_t)4096 * 800;        // 4096*128
  float* w1p = h3 + (size_t)4096 * 128;        // 676*16*12
  float* w2p = w1p + (size_t)676 * 16 * 12;    // 121*32*144
  float* f1T = w2p + (size_t)121 * 32 * 144;   // 128*800
  float* f2p = f1T + (size_t)128 * 800;        // 16*128

  // Weight repack prologue (tiny; amortized 128x by batch tiles)
  k_pack_w1 <<<(676 * 16 * 12 + 255) / 256, 256, 0, stream>>>(w1, w1p);
  k_pack_w2 <<<(121 * 32 * 144 + 255) / 256, 256, 0, stream>>>(w2, w2p);
  k_pack_fc1<<<(128 * 800 + 255) / 256, 256, 0, stream>>>(fc1_w, f1T);
  k_pack_fc2<<<(16 * 128 + 255) / 256, 256, 0, stream>>>(fc2_w, f2p);

  k_lc1<<<169 * 128 / 8, 256, 0, stream>>>(x, w1p, b1, h1);
  k_lc2<<< 25 * 128 / 8, 256, 0, stream>>>(h1, w2p, b2, h2);
  k_fc1<<<  8 * 128 / 8, 256, 0, stream>>>(h2, f1T, fc1_b, h3);
  k_fc2<<<      128 / 8, 256, 0, stream>>>(h3, f2p, fc2_b, (float*)d_out);
}